// TransposedAttentionBlock_20306605376065
// MI455X (gfx1250) — compile-verified
//
#include <hip/hip_runtime.h>
#include <cstddef>
#include <cstdint>

// ---------------------------------------------------------------------------
// Transposed (channel) attention block, fp32, CDNA5.
//  - v_wmma_f32_16x16x4_f32 for all GEMMs (exact fp32 semantics)
//  - double-buffered global_load_async_to_lds_b128/b32 staging (GVS form:
//    SGPR base + hoisted per-lane 32-bit offsets), ASYNCcnt pipelining
//  - algebra: out = x + E_b @ x + d_b with E_b = Wproj@attn_b@Wv (BN folded
//    into QKV weights, L2-norms folded into Gram epilogue). v/fused never
//    materialized. Workspace ~232 MB.
// ---------------------------------------------------------------------------

typedef float v2f __attribute__((ext_vector_type(2)));
typedef float v8f __attribute__((ext_vector_type(8)));

#define B_   16
#define C_   384
#define C2_  768
#define C3_  1152
#define N_   4096

#define TMB  128      // block tile M
#define TNB  64       // block tile N
#define KP   32       // K panel
#define KST  36       // padded LDS row stride (16B-aligned rows, conflict-free)

// workspace float offsets
#define OFF_QK    ((size_t)0)
#define SZ_QK     ((size_t)B_ * C2_ * N_)                  // 50,331,648
#define OFF_WEFF  (OFF_QK + SZ_QK)
#define OFF_BEFF  (OFF_WEFF + (size_t)C3_ * C_)
#define OFF_SCALE (OFF_BEFF + C3_)
#define OFF_SHIFT (OFF_SCALE + C_)
#define OFF_INVQ  (OFF_SHIFT + C_)
#define OFF_INVK  (OFF_INVQ + (size_t)B_ * C_)
#define OFF_ATTN  (OFF_INVK + (size_t)B_ * C_)
#define OFF_T1    (OFF_ATTN + (size_t)B_ * C_ * C_)
#define OFF_E     (OFF_T1 + (size_t)B_ * C_ * C_)
#define OFF_CV    (OFF_E + (size_t)B_ * C_ * C_)
#define OFF_D     (OFF_CV + (size_t)B_ * C_)

// ---------------------------------------------------------------------------
// Async global->LDS copies, GVS addressing: mem = SGPR_base + VGPR_off.
// LDS byte offset = low 32 bits of the generic pointer (ISA 10.2 aperture).
// Tracked by ASYNCcnt; fenced with s_wait_asynccnt.
// ---------------------------------------------------------------------------
__device__ __forceinline__ void async_b128(unsigned lds, unsigned voff, unsigned long long sbase)
{
    asm volatile("global_load_async_to_lds_b128 %0, %1, %2"
                 :: "v"(lds), "v"(voff), "s"(sbase) : "memory");
}
__device__ __forceinline__ void async_b32(unsigned lds, unsigned voff, unsigned long long sbase)
{
    asm volatile("global_load_async_to_lds_b32 %0, %1, %2"
                 :: "v"(lds), "v"(voff), "s"(sbase) : "memory");
}
__device__ __forceinline__ void wait_async()
{
    asm volatile("s_wait_asynccnt 0x0" ::: "memory");
}

// ---------------------------------------------------------------------------
// Kernel 1: per-channel batch-norm statistics -> scale/shift
// ---------------------------------------------------------------------------
__global__ void bn_stats_kernel(const float* __restrict__ x,
                                const float* __restrict__ gamma,
                                const float* __restrict__ beta,
                                float* __restrict__ scale,
                                float* __restrict__ shift)
{
    const int c   = blockIdx.x;
    const int tid = threadIdx.x;
    __shared__ float rs[256], rs2[256];
    float s = 0.f, s2 = 0.f;
    const float* xc = x + (size_t)c * N_;
    for (int i = tid; i < B_ * N_; i += 256) {
        int b = i >> 12, n = i & (N_ - 1);
        float v = xc[(size_t)b * C_ * N_ + n];
        s += v; s2 += v * v;
    }
    rs[tid] = s; rs2[tid] = s2; __syncthreads();
    for (int st = 128; st > 0; st >>= 1) {
        if (tid < st) { rs[tid] += rs[tid + st]; rs2[tid] += rs2[tid + st]; }
        __syncthreads();
    }
    if (tid == 0) {
        const float inv = 1.0f / (float)(B_ * N_);
        float mean = rs[0] * inv;
        float var  = rs2[0] * inv - mean * mean;
        float sc   = gamma[c] * rsqrtf(var + 1e-5f);
        scale[c]   = sc;
        shift[c]   = beta[c] - mean * sc;
    }
}

// ---------------------------------------------------------------------------
// Kernel 2: fold BN into QKV weights:  w'[o][c]=w*scale[c], b'[o]=b+sum w*shift
// ---------------------------------------------------------------------------
__global__ void fold_weights_kernel(const float* __restrict__ wqkv,
                                    const float* __restrict__ bqkv,
                                    const float* __restrict__ scale,
                                    const float* __restrict__ shift,
                                    float* __restrict__ weff,
                                    float* __restrict__ beff)
{
    const int o   = blockIdx.x;
    const int tid = threadIdx.x;
    __shared__ float red[256];
    float bs = 0.f;
    for (int c = tid; c < C_; c += 256) {
        float w = wqkv[(size_t)o * C_ + c];
        weff[(size_t)o * C_ + c] = w * scale[c];
        bs += w * shift[c];
    }
    red[tid] = bs; __syncthreads();
    for (int st = 128; st > 0; st >>= 1) {
        if (tid < st) red[tid] += red[tid + st];
        __syncthreads();
    }
    if (tid == 0) beff[o] = bqkv[o] + red[0];
}

// ---------------------------------------------------------------------------
// Stage one K-panel pair (A: TMB x KP, B: TNB x KP) into LDS, async.
// BT=true : B source is [N][K] row-major (K contiguous)  -> async b128 copy
// BT=false: B source is [K][N] row-major (N contiguous)  -> async b32 scatter
//           (transpose happens in the per-lane LDS destination address)
// Per-lane VGPR offsets are kp-invariant -> hoisted; base advances via SGPR.
// ---------------------------------------------------------------------------
template<bool BT>
__device__ __forceinline__ void stage_panels(float* As, float* Bs,
                                             const float* __restrict__ A, int lda,
                                             const float* __restrict__ Bsrc, int ldb,
                                             int kp)
{
    const int tid = threadIdx.x;
    {   // A panel
        unsigned voff = (unsigned)((((tid >> 3) * lda) + ((tid & 7) << 2)) * 4);
        unsigned lds0 = (unsigned)(uintptr_t)(As + (tid >> 3) * KST + ((tid & 7) << 2));
        unsigned long long sb = (unsigned long long)(uintptr_t)(A + kp);
#pragma unroll
        for (int i = 0; i < TMB / 32; i++)
            async_b128(lds0 + (unsigned)(i * 32 * KST * 4),
                       voff + (unsigned)(i * 32 * lda * 4), sb);
    }
    if (BT) {
        unsigned voff = (unsigned)((((tid >> 3) * ldb) + ((tid & 7) << 2)) * 4);
        unsigned lds0 = (unsigned)(uintptr_t)(Bs + (tid >> 3) * KST + ((tid & 7) << 2));
        unsigned long long sb = (unsigned long long)(uintptr_t)(Bsrc + kp);
#pragma unroll
        for (int i = 0; i < TNB / 32; i++)
            async_b128(lds0 + (unsigned)(i * 32 * KST * 4),
                       voff + (unsigned)(i * 32 * ldb * 4), sb);
    } else {
        unsigned voff = (unsigned)((((tid >> 6) * ldb) + (tid & 63)) * 4);
        unsigned lds0 = (unsigned)(uintptr_t)(Bs + (tid & 63) * KST + (tid >> 6));
        unsigned long long sb = (unsigned long long)(uintptr_t)(Bsrc + (size_t)kp * ldb);
#pragma unroll
        for (int i = 0; i < 8; i++)
            async_b32(lds0 + (unsigned)(i * 16),
                      voff + (unsigned)(i * 16 * ldb), sb);
    }
}

// ---------------------------------------------------------------------------
// One K-panel of WMMAs (8 k-steps x 4 wmma) from given LDS fragment pointers.
// ---------------------------------------------------------------------------
__device__ __forceinline__ void mma_panel(const float* aF0, const float* aF1,
                                          const float* bF0, const float* bF1,
                                          v8f (&acc)[2][2])
{
#pragma unroll
    for (int k0 = 0; k0 < KP; k0 += 4) {
        v2f a0 = *(const v2f*)(aF0 + k0);
        v2f a1 = *(const v2f*)(aF1 + k0);
        v2f b0 = *(const v2f*)(bF0 + k0);
        v2f b1 = *(const v2f*)(bF1 + k0);
        acc[0][0] = __builtin_amdgcn_wmma_f32_16x16x4_f32(false, a0, false, b0, (short)0, acc[0][0], false, false);
        acc[0][1] = __builtin_amdgcn_wmma_f32_16x16x4_f32(false, a0, false, b1, (short)0, acc[0][1], false, false);
        acc[1][0] = __builtin_amdgcn_wmma_f32_16x16x4_f32(false, a1, false, b0, (short)0, acc[1][0], false, false);
        acc[1][1] = __builtin_amdgcn_wmma_f32_16x16x4_f32(false, a1, false, b1, (short)0, acc[1][1], false, false);
    }
}

// ---------------------------------------------------------------------------
// 128x64 block GEMM core: double-buffered async staging + fp32 WMMA.
// 8 waves, each wave owns a 32x32 output (4x v8f). Ktot/KP must be even
// (holds for K=384 and K=4096).
// ---------------------------------------------------------------------------
template<bool BT>
__device__ __forceinline__ void gemm_128x64(const float* __restrict__ A, int lda,
                                            const float* __restrict__ Bsrc, int ldb,
                                            int Ktot,
                                            float* As0, float* Bs0,
                                            float* As1, float* Bs1,
                                            v8f (&acc)[2][2])
{
    const int lane = threadIdx.x & 31;
    const int wave = threadIdx.x >> 5;
    const int wm   = wave & 3;
    const int wn   = wave >> 2;
    const int half = lane >> 4;
    const int mi   = lane & 15;

    acc[0][0] = {}; acc[0][1] = {}; acc[1][0] = {}; acc[1][1] = {};

    const float* aF0_0 = As0 + (wm * 32 + mi) * KST + 2 * half;
    const float* aF1_0 = aF0_0 + 16 * KST;
    const float* bF0_0 = Bs0 + (wn * 32 + mi) * KST + 2 * half;
    const float* bF1_0 = bF0_0 + 16 * KST;
    const float* aF0_1 = As1 + (wm * 32 + mi) * KST + 2 * half;
    const float* aF1_1 = aF0_1 + 16 * KST;
    const float* bF0_1 = Bs1 + (wn * 32 + mi) * KST + 2 * half;
    const float* bF1_1 = bF0_1 + 16 * KST;

    stage_panels<BT>(As0, Bs0, A, lda, Bsrc, ldb, 0);

    for (int kp = 0; kp < Ktot; kp += 2 * KP) {
        wait_async();                 // panel kp (buf0) landed
        __syncthreads();              // visible to all; buf1 readers (kp-KP) done
        if (kp + KP < Ktot)
            stage_panels<BT>(As1, Bs1, A, lda, Bsrc, ldb, kp + KP);
        mma_panel(aF0_0, aF1_0, bF0_0, bF1_0, acc);

        wait_async();                 // panel kp+KP (buf1) landed
        __syncthreads();              // buf0 readers done before overwrite
        if (kp + 2 * KP < Ktot)
            stage_panels<BT>(As0, Bs0, A, lda, Bsrc, ldb, kp + 2 * KP);
        mma_panel(aF0_1, aF1_1, bF0_1, bF1_1, acc);
    }
}

#define GEMM_SHARED                                          \
    __shared__ __align__(16) float As0[TMB * KST];           \
    __shared__ __align__(16) float Bs0[TNB * KST];           \
    __shared__ __align__(16) float As1[TMB * KST];           \
    __shared__ __align__(16) float Bs1[TNB * KST];

// ---------------------------------------------------------------------------
// Kernel 3: qk[b] = w'[0:768] (768x384) @ x[b] (384x4096) + b'[0:768]
// ---------------------------------------------------------------------------
__global__ void __launch_bounds__(256)
qk_gemm_kernel(const float* __restrict__ x,
               const float* __restrict__ weff,
               const float* __restrict__ beff,
               float* __restrict__ qk)
{
    GEMM_SHARED
    const int b  = blockIdx.z;
    const int M0 = blockIdx.y * TMB;
    const int N0 = blockIdx.x * TNB;

    v8f acc[2][2];
    gemm_128x64<false>(weff + (size_t)M0 * C_, C_,
                       x + (size_t)b * C_ * N_ + N0, N_,
                       C_, As0, Bs0, As1, Bs1, acc);

    const int lane = threadIdx.x & 31, wave = threadIdx.x >> 5;
    const int wm = wave & 3, wn = wave >> 2, half = lane >> 4, mi = lane & 15;
    float* out = qk + (size_t)b * C2_ * N_;
#pragma unroll
    for (int tm = 0; tm < 2; tm++)
#pragma unroll
        for (int tn = 0; tn < 2; tn++) {
            int row = M0 + wm * 32 + tm * 16 + half * 8;
            int col = N0 + wn * 32 + tn * 16 + mi;
#pragma unroll
            for (int r = 0; r < 8; r++)
                out[(size_t)(row + r) * N_ + col] = acc[tm][tn][r] + beff[row + r];
        }
}

// ---------------------------------------------------------------------------
// Kernel 4: reciprocal L2 norms of q and k rows (over N)
// ---------------------------------------------------------------------------
__global__ void norms_kernel(const float* __restrict__ qk,
                             float* __restrict__ invq,
                             float* __restrict__ invk)
{
    const int c = blockIdx.x, b = blockIdx.y, tid = threadIdx.x;
    __shared__ float rq[256], rk[256];
    const float* qrow = qk + (size_t)b * C2_ * N_ + (size_t)c * N_;
    const float* krow = qrow + (size_t)C_ * N_;
    float sq = 0.f, sk = 0.f;
    for (int n = tid; n < N_; n += 256) {
        float a = qrow[n]; sq += a * a;
        float d = krow[n]; sk += d * d;
    }
    rq[tid] = sq; rk[tid] = sk; __syncthreads();
    for (int st = 128; st > 0; st >>= 1) {
        if (tid < st) { rq[tid] += rq[tid + st]; rk[tid] += rk[tid + st]; }
        __syncthreads();
    }
    if (tid == 0) {
        invq[b * C_ + c] = 1.0f / fmaxf(sqrtf(rq[0]), 1e-12f);
        invk[b * C_ + c] = 1.0f / fmaxf(sqrtf(rk[0]), 1e-12f);
    }
}

// ---------------------------------------------------------------------------
// Kernel 5: Gram G[b] = q[b] @ k[b]^T  (K = N = 4096), scaled by
// temperature/(|q||k|) in the epilogue.
// ---------------------------------------------------------------------------
__global__ void __launch_bounds__(256)
gram_kernel(const float* __restrict__ qk,
            const float* __restrict__ invq,
            const float* __restrict__ invk,
            const float* __restrict__ temperature,
            float* __restrict__ attn)
{
    GEMM_SHARED
    const int b  = blockIdx.z;
    const int M0 = blockIdx.y * TMB;
    const int N0 = blockIdx.x * TNB;
    const float* qb = qk + (size_t)b * C2_ * N_;
    const float* kb = qb + (size_t)C_ * N_;

    v8f acc[2][2];
    gemm_128x64<true>(qb + (size_t)M0 * N_, N_,
                      kb + (size_t)N0 * N_, N_,
                      N_, As0, Bs0, As1, Bs1, acc);

    const int lane = threadIdx.x & 31, wave = threadIdx.x >> 5;
    const int wm = wave & 3, wn = wave >> 2, half = lane >> 4, mi = lane & 15;
    const float t = temperature[0];
    const float* iqv = invq + b * C_;
    const float* ikv = invk + b * C_;
    float* outp = attn + (size_t)b * C_ * C_;
#pragma unroll
    for (int tm = 0; tm < 2; tm++)
#pragma unroll
        for (int tn = 0; tn < 2; tn++) {
            int row = M0 + wm * 32 + tm * 16 + half * 8;
            int col = N0 + wn * 32 + tn * 16 + mi;
            float ik = ikv[col] * t;
#pragma unroll
            for (int r = 0; r < 8; r++)
                outp[(size_t)(row + r) * C_ + col] = acc[tm][tn][r] * iqv[row + r] * ik;
        }
}

// ---------------------------------------------------------------------------
// Kernel 6: row softmax over 384 entries (128 threads, 3 elems/thread)
// ---------------------------------------------------------------------------
__global__ void softmax_kernel(float* __restrict__ attn)
{
    const int row = blockIdx.x;            // 0 .. B*C-1
    const int tid = threadIdx.x;
    float* p = attn + (size_t)row * C_;
    __shared__ float red[128];
    float v0 = p[tid], v1 = p[tid + 128], v2 = p[tid + 256];
    float m = fmaxf(fmaxf(v0, v1), v2);
    red[tid] = m; __syncthreads();
    for (int st = 64; st > 0; st >>= 1) {
        if (tid < st) red[tid] = fmaxf(red[tid], red[tid + st]);
        __syncthreads();
    }
    m = red[0]; __syncthreads();
    v0 = __expf(v0 - m); v1 = __expf(v1 - m); v2 = __expf(v2 - m);
    red[tid] = v0 + v1 + v2; __syncthreads();
    for (int st = 64; st > 0; st >>= 1) {
        if (tid < st) red[tid] += red[tid + st];
        __syncthreads();
    }
    float inv = 1.0f / red[0];
    p[tid] = v0 * inv; p[tid + 128] = v1 * inv; p[tid + 256] = v2 * inv;
}

// ---------------------------------------------------------------------------
// Kernel 7: T1[b] = attn[b] (384x384) @ Wv' (384x384)
// ---------------------------------------------------------------------------
__global__ void __launch_bounds__(256)
t1_gemm_kernel(const float* __restrict__ attn,
               const float* __restrict__ weff,
               float* __restrict__ T1)
{
    GEMM_SHARED
    const int b  = blockIdx.z;
    const int M0 = blockIdx.y * TMB;
    const int N0 = blockIdx.x * TNB;

    v8f acc[2][2];
    gemm_128x64<false>(attn + (size_t)b * C_ * C_ + (size_t)M0 * C_, C_,
                       weff + (size_t)C2_ * C_ + N0, C_,
                       C_, As0, Bs0, As1, Bs1, acc);

    const int lane = threadIdx.x & 31, wave = threadIdx.x >> 5;
    const int wm = wave & 3, wn = wave >> 2, half = lane >> 4, mi = lane & 15;
    float* out = T1 + (size_t)b * C_ * C_;
#pragma unroll
    for (int tm = 0; tm < 2; tm++)
#pragma unroll
        for (int tn = 0; tn < 2; tn++) {
            int row = M0 + wm * 32 + tm * 16 + half * 8;
            int col = N0 + wn * 32 + tn * 16 + mi;
#pragma unroll
            for (int r = 0; r < 8; r++)
                out[(size_t)(row + r) * C_ + col] = acc[tm][tn][r];
        }
}

// ---------------------------------------------------------------------------
// Kernel 8: cv[b][m] = sum_o attn[b][m][o] * bv'[o]
// ---------------------------------------------------------------------------
__global__ void cv_kernel(const float* __restrict__ attn,
                          const float* __restrict__ beff,
                          float* __restrict__ cv)
{
    const int row = blockIdx.x;            // b*C + m
    const int tid = threadIdx.x;
    __shared__ float red[128];
    const float* a  = attn + (size_t)row * C_;
    const float* bv = beff + C2_;
    float s = a[tid] * bv[tid] + a[tid + 128] * bv[tid + 128] + a[tid + 256] * bv[tid + 256];
    red[tid] = s; __syncthreads();
    for (int st = 64; st > 0; st >>= 1) {
        if (tid < st) red[tid] += red[tid + st];
        __syncthreads();
    }
    if (tid == 0) cv[row] = red[0];
}

// ---------------------------------------------------------------------------
// Kernel 9: E[b] = Wproj (384x384) @ T1[b] (384x384)
// ---------------------------------------------------------------------------
__global__ void __launch_bounds__(256)
e_gemm_kernel(const float* __restrict__ wproj,
              const float* __restrict__ T1,
              float* __restrict__ E)
{
    GEMM_SHARED
    const int b  = blockIdx.z;
    const int M0 = blockIdx.y * TMB;
    const int N0 = blockIdx.x * TNB;

    v8f acc[2][2];
    gemm_128x64<false>(wproj + (size_t)M0 * C_, C_,
                       T1 + (size_t)b * C_ * C_ + N0, C_,
                       C_, As0, Bs0, As1, Bs1, acc);

    const int lane = threadIdx.x & 31, wave = threadIdx.x >> 5;
    const int wm = wave & 3, wn = wave >> 2, half = lane >> 4, mi = lane & 15;
    float* out = E + (size_t)b * C_ * C_;
#pragma unroll
    for (int tm = 0; tm < 2; tm++)
#pragma unroll
        for (int tn = 0; tn < 2; tn++) {
            int row = M0 + wm * 32 + tm * 16 + half * 8;
            int col = N0 + wn * 32 + tn * 16 + mi;
#pragma unroll
            for (int r = 0; r < 8; r++)
                out[(size_t)(row + r) * C_ + col] = acc[tm][tn][r];
        }
}

// ---------------------------------------------------------------------------
// Kernel 10: d[b][p] = sum_m wproj[p][m]*cv[b][m] + bproj[p]
// ---------------------------------------------------------------------------
__global__ void d_kernel(const float* __restrict__ wproj,
                         const float* __restrict__ bproj,
                         const float* __restrict__ cv,
                         float* __restrict__ dvec)
{
    const int p = blockIdx.x, b = blockIdx.y, tid = threadIdx.x;
    __shared__ float red[128];
    const float* w = wproj + (size_t)p * C_;
    const float* c = cv + (size_t)b * C_;
    float s = w[tid] * c[tid] + w[tid + 128] * c[tid + 128] + w[tid + 256] * c[tid + 256];
    red[tid] = s; __syncthreads();
    for (int st = 64; st > 0; st >>= 1) {
        if (tid < st) red[tid] += red[tid + st];
        __syncthreads();
    }
    if (tid == 0) dvec[b * C_ + p] = red[0] + bproj[p];
}

// ---------------------------------------------------------------------------
// Kernel 11: out[b] = E[b] @ x[b] + d[b] + x[b]   (residual)
// ---------------------------------------------------------------------------
__global__ void __launch_bounds__(256)
final_gemm_kernel(const float* __restrict__ E,
                  const float* __restrict__ dvec,
                  const float* __restrict__ x,
                  float* __restrict__ out)
{
    GEMM_SHARED
    const int b  = blockIdx.z;
    const int M0 = blockIdx.y * TMB;
    const int N0 = blockIdx.x * TNB;

    v8f acc[2][2];
    gemm_128x64<false>(E + (size_t)b * C_ * C_ + (size_t)M0 * C_, C_,
                       x + (size_t)b * C_ * N_ + N0, N_,
                       C_, As0, Bs0, As1, Bs1, acc);

    const int lane = threadIdx.x & 31, wave = threadIdx.x >> 5;
    const int wm = wave & 3, wn = wave >> 2, half = lane >> 4, mi = lane & 15;
    const float* xb = x + (size_t)b * C_ * N_;
    const float* db = dvec + (size_t)b * C_;
    float* ob = out + (size_t)b * C_ * N_;
#pragma unroll
    for (int tm = 0; tm < 2; tm++)
#pragma unroll
        for (int tn = 0; tn < 2; tn++) {
            int row = M0 + wm * 32 + tm * 16 + half * 8;
            int col = N0 + wn * 32 + tn * 16 + mi;
#pragma unroll
            for (int r = 0; r < 8; r++) {
                size_t idx = (size_t)(row + r) * N_ + col;
                ob[idx] = acc[tm][tn][r] + db[row + r] + xb[idx];
            }
        }
}

// ---------------------------------------------------------------------------
extern "C" void kernel_launch(void* const* d_in, const int* in_sizes, int n_in,
                              void* d_out, int out_size, void* d_ws, size_t ws_size,
                              hipStream_t stream)
{
    (void)in_sizes; (void)n_in; (void)out_size; (void)ws_size;
    const float* x     = (const float*)d_in[0];
    const float* gamma = (const float*)d_in[1];
    const float* beta  = (const float*)d_in[2];
    const float* wqkv  = (const float*)d_in[3];
    const float* bqkv  = (const float*)d_in[4];
    const float* wproj = (const float*)d_in[5];
    const float* bproj = (const float*)d_in[6];
    const float* temp  = (const float*)d_in[7];
    float* out = (float*)d_out;
    float* ws  = (float*)d_ws;

    float* qk    = ws + OFF_QK;
    float* weff  = ws + OFF_WEFF;
    float* beff  = ws + OFF_BEFF;
    float* scale = ws + OFF_SCALE;
    float* shift = ws + OFF_SHIFT;
    float* invq  = ws + OFF_INVQ;
    float* invk  = ws + OFF_INVK;
    float* attn  = ws + OFF_ATTN;
    float* T1    = ws + OFF_T1;
    float* E     = ws + OFF_E;
    float* cv    = ws + OFF_CV;
    float* dvec  = ws + OFF_D;

    bn_stats_kernel<<<C_, 256, 0, stream>>>(x, gamma, beta, scale, shift);
    fold_weights_kernel<<<C3_, 256, 0, stream>>>(wqkv, bqkv, scale, shift, weff, beff);
    qk_gemm_kernel<<<dim3(N_ / TNB, C2_ / TMB, B_), 256, 0, stream>>>(x, weff, beff, qk);
    norms_kernel<<<dim3(C_, B_), 256, 0, stream>>>(qk, invq, invk);
    gram_kernel<<<dim3(C_ / TNB, C_ / TMB, B_), 256, 0, stream>>>(qk, invq, invk, temp, attn);
    softmax_kernel<<<B_ * C_, 128, 0, stream>>>(attn);
    t1_gemm_kernel<<<dim3(C_ / TNB, C_ / TMB, B_), 256, 0, stream>>>(attn, weff, T1);
    cv_kernel<<<B_ * C_, 128, 0, stream>>>(attn, beff, cv);
    e_gemm_kernel<<<dim3(C_ / TNB, C_ / TMB, B_), 256, 0, stream>>>(wproj, T1, E);
    d_kernel<<<dim3(C_, B_), 128, 0, stream>>>(wproj, bproj, cv, dvec);
    final_gemm_kernel<<<dim3(N_ / TNB, C_ / TMB, B_), 256, 0, stream>>>(E, dvec, x, out);
}